// GCN_26792005992869
// MI455X (gfx1250) — compile-verified
//
#include <hip/hip_runtime.h>
#include <math.h>

// Problem constants (match reference)
#define NN  50000          // nodes
#define NE  800000         // edges
#define KIN 512            // input feature dim
#define HD  256            // hidden dim
#define NF  (NN * HD)      // elements in a node-feature matrix

typedef __attribute__((ext_vector_type(2))) float v2f;
typedef __attribute__((ext_vector_type(8))) float v8f;

// ---------------------------------------------------------------------------
// Utility: vectorized fill
// ---------------------------------------------------------------------------
__global__ void fill_f4(float4* __restrict__ p, int n4, float v) {
    int i = blockIdx.x * blockDim.x + threadIdx.x;
    if (i < n4) p[i] = make_float4(v, v, v, v);
}

// ---------------------------------------------------------------------------
// Degree accumulation (float atomics) and norm finalize: 1/sqrt(max(deg,1))
// ---------------------------------------------------------------------------
__global__ void degree_kernel(const int* __restrict__ src, const int* __restrict__ dst,
                              float* __restrict__ outdeg, float* __restrict__ indeg) {
    int e = blockIdx.x * blockDim.x + threadIdx.x;
    if (e < NE) {
        atomicAdd(outdeg + src[e], 1.0f);
        atomicAdd(indeg + dst[e], 1.0f);
    }
}

__global__ void norm_finalize(float* __restrict__ d, int n) {
    int i = blockIdx.x * blockDim.x + threadIdx.x;
    if (i < n) d[i] = 1.0f / sqrtf(fmaxf(d[i], 1.0f));
}

// ---------------------------------------------------------------------------
// hs[node, :] = h[node, :] * norm_src[node]   (float4 vectorized)
// ---------------------------------------------------------------------------
__global__ void scale_rows(const float* __restrict__ h, const float* __restrict__ ns,
                           float* __restrict__ hs) {
    int i = blockIdx.x * blockDim.x + threadIdx.x;   // float4 index
    if (i < NF / 4) {
        int node = i / (HD / 4);
        float s = ns[node];
        float4 v = ((const float4*)h)[i];
        v.x *= s; v.y *= s; v.z *= s; v.w *= s;
        ((float4*)hs)[i] = v;
    }
}

// ---------------------------------------------------------------------------
// Edge gather + scatter-add: agg[dst[e], :] += hs[src[e], :]
// One wave (32 lanes) per edge; 2x float4 per lane covers 256 floats.
// hs and agg (51 MB each) are L2-resident (192 MB L2): the gather and the
// atomic adds resolve in L2, not HBM.
// ---------------------------------------------------------------------------
__global__ void scatter_edges(const float* __restrict__ hs, const int* __restrict__ src,
                              const int* __restrict__ dst, float* __restrict__ agg) {
    int t = threadIdx.x;
    int e = blockIdx.x * 8 + (t >> 5);   // 8 waves per 256-thread block
    if (e >= NE) return;
    int lane = t & 31;
    int s = src[e];
    int d = dst[e];
    const float4* hp = (const float4*)(hs + (size_t)s * HD);
    float* ap = agg + (size_t)d * HD;
#pragma unroll
    for (int j = 0; j < 2; ++j) {
        int u = j * 32 + lane;           // float4 unit 0..63 within row
        float4 v = hp[u];
        atomicAdd(ap + u * 4 + 0, v.x);
        atomicAdd(ap + u * 4 + 1, v.y);
        atomicAdd(ap + u * 4 + 2, v.z);
        atomicAdd(ap + u * 4 + 3, v.w);
    }
}

// ---------------------------------------------------------------------------
// C[M x 256] = relu( (A[M x K] @ W[K x 256]) * rowscale[m] + bias[n] )
// rowscale == nullptr -> no scaling (encoder).
// Block: 256 threads (8 wave32), 16 rows x full 256 cols per block.
// K staged through LDS in 32-wide slabs; fp32 WMMA (16x16x4) accumulate.
// B slab is stored TRANSPOSED in LDS ([N][K], even stride 34) so each lane's
// (K,K+1) operand pair is one aligned 64-bit DS load, and reads are
// bank-conflict-free ({34n mod 64} distinct, +2 disjoint for lane half 1).
// M (=50000) is an exact multiple of 16 and K in {512,256} of 32 -> no guards.
// ---------------------------------------------------------------------------
__global__ __launch_bounds__(256)
void gemm_relu(const float* __restrict__ A, const float* __restrict__ W,
               const float* __restrict__ bias, const float* __restrict__ rowscale,
               float* __restrict__ C, int K) {
    __shared__ float lA[16][33];     // [M][Kslab], pad 33 (scalar/2addr reads)
    __shared__ float lB[256][34];    // [N][Kslab], even pad 34 (b64 reads)

    const int t    = threadIdx.x;
    const int m0   = blockIdx.x << 4;   // 16 rows per block
    const int lane = t & 31;
    const int wv   = t >> 5;            // wave id 0..7
    const int tn   = wv << 5;           // this wave's N base (2 tiles of 16)
    const int nlo  = lane & 15;         // lane's M (for A) and N (for B/C)
    const int half = lane >> 4;         // lane half selects K pair / M+8
    const int kb   = half << 1;         // K offset within 4-step: 0 or 2

    v8f acc0 = {};
    v8f acc1 = {};

    for (int k0 = 0; k0 < K; k0 += 32) {
        // --- stage A tile: 16x32, one float2 per thread, coalesced per row ---
        {
            int r = t >> 4;             // 0..15
            int c = (t & 15) << 1;      // 0,2,...,30
            const float* gp = A + (size_t)(m0 + r) * K + (k0 + c);
            lA[r][c]     = gp[0];
            lA[r][c + 1] = gp[1];
        }
        // --- stage B slab transposed: read W rows coalesced, write [N][K] ---
        {
            int c  = (t & 63) << 2;     // N base 0..252
            int rb = t >> 6;            // 0..3
#pragma unroll
            for (int i = 0; i < 8; ++i) {
                int r = rb + (i << 2);  // K row 0..31
                const float4 w4 = *(const float4*)(W + (size_t)(k0 + r) * HD + c);
                lB[c + 0][r] = w4.x;
                lB[c + 1][r] = w4.y;
                lB[c + 2][r] = w4.z;
                lB[c + 3][r] = w4.w;
            }
        }
        // Prefetch next K slab while this one is consumed (global_prefetch_b8)
        if (k0 + 32 < K) {
            __builtin_prefetch(A + (size_t)(m0 + (t >> 4)) * K + (k0 + 32 + ((t & 15) << 1)), 0, 1);
            __builtin_prefetch(W + (size_t)(k0 + 32 + (t >> 6)) * HD + ((t & 63) << 2), 0, 1);
        }
        __syncthreads();

        // --- fp32 WMMA over the 32-wide K slab, 4 at a time ---
#pragma unroll
        for (int kk = 0; kk < 32; kk += 4) {
            v2f a;
            // A 16x4: lane = M, VGPR0/1 = K{kb, kb+1}
            a.x = lA[nlo][kk + kb];
            a.y = lA[nlo][kk + kb + 1];
            // B 4x16: lane = N; transposed LDS row gives aligned 64-bit pair
            v2f b0 = *(const v2f*)&lB[tn + nlo][kk + kb];
            v2f b1 = *(const v2f*)&lB[tn + 16 + nlo][kk + kb];
            acc0 = __builtin_amdgcn_wmma_f32_16x16x4_f32(
                false, a, false, b0, (short)0, acc0, false, false);
            acc1 = __builtin_amdgcn_wmma_f32_16x16x4_f32(
                false, a, false, b1, (short)0, acc1, false, false);
        }
        __syncthreads();
    }

    // --- epilogue: C/D layout -> VGPR j: M = j + 8*half, N = nlo ---
    const float bia0 = bias[tn + nlo];
    const float bia1 = bias[tn + 16 + nlo];
#pragma unroll
    for (int j = 0; j < 8; ++j) {
        int m = m0 + j + (half << 3);
        float rs = rowscale ? rowscale[m] : 1.0f;
        float v0 = fmaxf(acc0[j] * rs + bia0, 0.0f);
        float v1 = fmaxf(acc1[j] * rs + bia1, 0.0f);
        C[(size_t)m * HD + tn + nlo]      = v0;
        C[(size_t)m * HD + tn + 16 + nlo] = v1;
    }
}

// ---------------------------------------------------------------------------
// Orchestration
// ---------------------------------------------------------------------------
extern "C" void kernel_launch(void* const* d_in, const int* in_sizes, int n_in,
                              void* d_out, int out_size, void* d_ws, size_t ws_size,
                              hipStream_t stream) {
    const float* x     = (const float*)d_in[0];   // [NN, KIN]
    const int*   esrc  = (const int*)  d_in[1];   // [NE]
    const int*   edst  = (const int*)  d_in[2];   // [NE]
    const float* encW  = (const float*)d_in[3];   // [KIN, HD]
    const float* encb  = (const float*)d_in[4];   // [HD]
    const float* convW = (const float*)d_in[5];   // [3, HD, HD]
    const float* convb = (const float*)d_in[6];   // [3, HD]
    float* out = (float*)d_out;                   // [NN, HD] — also serves as h

    char* ws = (char*)d_ws;
    float* hs   = (float*)(ws);                                    // 51.2 MB
    float* agg  = (float*)(ws + (size_t)NF * 4);                   // 51.2 MB
    float* nsrc = (float*)(ws + (size_t)NF * 8);                   // 200 KB
    float* ndst = (float*)(ws + (size_t)NF * 8 + (size_t)NN * 4);  // 200 KB

    // Degree norms
    fill_f4<<<(NN / 4 + 255) / 256, 256, 0, stream>>>((float4*)nsrc, NN / 4, 0.0f);
    fill_f4<<<(NN / 4 + 255) / 256, 256, 0, stream>>>((float4*)ndst, NN / 4, 0.0f);
    degree_kernel<<<(NE + 255) / 256, 256, 0, stream>>>(esrc, edst, nsrc, ndst);
    norm_finalize<<<(NN + 255) / 256, 256, 0, stream>>>(nsrc, NN);
    norm_finalize<<<(NN + 255) / 256, 256, 0, stream>>>(ndst, NN);

    // Encoder: h = relu(x @ encW + encb)  -> d_out
    gemm_relu<<<NN / 16, 256, 0, stream>>>(x, encW, encb, nullptr, out, KIN);

    // 3 GraphConv layers
    for (int i = 0; i < 3; ++i) {
        scale_rows<<<(NF / 4 + 255) / 256, 256, 0, stream>>>(out, nsrc, hs);
        fill_f4<<<(NF / 4 + 255) / 256, 256, 0, stream>>>((float4*)agg, NF / 4, 0.0f);
        scatter_edges<<<(NE + 7) / 8, 256, 0, stream>>>(hs, esrc, edst, agg);
        gemm_relu<<<NN / 16, 256, 0, stream>>>(agg, convW + (size_t)i * HD * HD,
                                               convb + (size_t)i * HD, ndst, out, HD);
    }
}